// baseline_net_81054622810539
// MI455X (gfx1250) — compile-verified
//
#include <hip/hip_runtime.h>
#include <hip/hip_bf16.h>

typedef __attribute__((ext_vector_type(2))) float v2f;
typedef __attribute__((ext_vector_type(8))) float v8f;

#define N_FEAT 128
#define HIDDEN 64
#define N_OUT  21

// ---------------- utility: fill ----------------
__global__ void fill_kernel(float* __restrict__ p, float v, int n) {
    int i = blockIdx.x * blockDim.x + threadIdx.x;
    if (i < n) p[i] = v;
}

// ---------------- degree accumulation (deg pre-set to 1.0 for self loop) ----------------
__global__ void deg_kernel(const int* __restrict__ dstI, float* __restrict__ deg, int E) {
    int e = blockIdx.x * blockDim.x + threadIdx.x;
    if (e < E) atomicAdd(&deg[dstI[e]], 1.0f);
}

// ---------------- deg -> deg^-1/2 in place ----------------
__global__ void rsqrt_kernel(float* __restrict__ deg, int n) {
    int i = blockIdx.x * blockDim.x + threadIdx.x;
    if (i < n) deg[i] = 1.0f / sqrtf(deg[i]);
}

// ---------------- GEMM1: h[N,64] = x[N,128] @ W1[128,64]  (WMMA f32 16x16x4) ----------------
// block = 128 threads (4 waves); each block does one 16-row M tile; wave w does cols [16w,16w+16)
__global__ void __launch_bounds__(128)
gemm1_kernel(const float* __restrict__ x, const float* __restrict__ W1,
             float* __restrict__ h, int N) {
    __shared__ float As[16 * N_FEAT];   // 8 KB A tile
    const int tid = threadIdx.x;
    const int m0  = blockIdx.x * 16;

    // cooperative load: A tile is contiguous (row stride == K == 128)
    const float4* gsrc = (const float4*)(x + (size_t)m0 * N_FEAT);
    float4* lds4 = (float4*)As;
    const int tileVec = 16 * N_FEAT / 4;            // 512 float4
    const int maxVec  = (int)(((size_t)N * N_FEAT - (size_t)m0 * N_FEAT) / 4);
    #pragma unroll
    for (int i = tid; i < tileVec; i += 128) {
        float4 z = make_float4(0.f, 0.f, 0.f, 0.f);
        lds4[i] = (i < maxVec) ? gsrc[i] : z;
    }
    __syncthreads();

    const int lane  = tid & 31;
    const int half  = lane >> 4;        // selects K pair {0,1} vs {2,3}
    const int idx16 = lane & 15;        // A row / B,D col
    const int n0    = (tid >> 5) * 16;  // wave -> N tile

    v8f acc = {};
    #pragma unroll
    for (int k0 = 0; k0 < N_FEAT; k0 += 4) {
        const int ka = k0 + half * 2;
        v2f a;
        a.x = As[idx16 * N_FEAT + ka];
        a.y = As[idx16 * N_FEAT + ka + 1];
        v2f b;
        b.x = W1[(size_t)ka       * HIDDEN + n0 + idx16];
        b.y = W1[(size_t)(ka + 1) * HIDDEN + n0 + idx16];
        acc = __builtin_amdgcn_wmma_f32_16x16x4_f32(false, a, false, b,
                                                    (short)0, acc, false, false);
    }
    #pragma unroll
    for (int r = 0; r < 8; ++r) {
        const int m = m0 + r + half * 8;
        if (m < N) h[(size_t)m * HIDDEN + n0 + idx16] = acc[r];
    }
}

// ---------------- layer-1 edge aggregation: agg[dst] += h[src] * dinv[src]*dinv[dst] ----------------
__global__ void agg1_kernel(const int* __restrict__ srcI, const int* __restrict__ dstI,
                            const float* __restrict__ dinv, const float* __restrict__ h,
                            float* __restrict__ agg, int E) {
    int t = blockIdx.x * 256 + threadIdx.x;   // (edge, feature) pair
    int e = t >> 6;
    int f = t & 63;
    if (e >= E) return;
    int s = srcI[e], d = dstI[e];
    float w = dinv[s] * dinv[d];
    atomicAdd(&agg[(size_t)d * HIDDEN + f], h[(size_t)s * HIDDEN + f] * w);
}

// ---------------- self loop + bias + ReLU (in place on agg) ----------------
__global__ void sbr_kernel(const float* __restrict__ h, const float* __restrict__ dinv,
                           const float* __restrict__ b1, float* __restrict__ agg, int total) {
    int i = blockIdx.x * blockDim.x + threadIdx.x;
    if (i >= total) return;
    int n = i >> 6;
    int f = i & 63;
    float dv = dinv[n];
    float v = agg[i] + h[i] * dv * dv + b1[f];
    agg[i] = v > 0.f ? v : 0.f;
}

// ---------------- GEMM2: h2[N,21] = hrelu[N,64] @ W2[64,21]  (WMMA f32 16x16x4) ----------------
// block = 64 threads (2 waves); wave 0 -> cols 0..15, wave 1 -> cols 16..20 (guarded)
__global__ void __launch_bounds__(64)
gemm2_kernel(const float* __restrict__ hrelu, const float* __restrict__ W2,
             float* __restrict__ h2, int N) {
    __shared__ float As[16 * HIDDEN];   // 4 KB
    const int tid = threadIdx.x;
    const int m0  = blockIdx.x * 16;

    const float4* gsrc = (const float4*)(hrelu + (size_t)m0 * HIDDEN);
    float4* lds4 = (float4*)As;
    const int tileVec = 16 * HIDDEN / 4;            // 256 float4
    const int maxVec  = (int)(((size_t)N * HIDDEN - (size_t)m0 * HIDDEN) / 4);
    #pragma unroll
    for (int i = tid; i < tileVec; i += 64) {
        float4 z = make_float4(0.f, 0.f, 0.f, 0.f);
        lds4[i] = (i < maxVec) ? gsrc[i] : z;
    }
    __syncthreads();

    const int lane  = tid & 31;
    const int half  = lane >> 4;
    const int idx16 = lane & 15;
    const int n0    = (tid >> 5) * 16;
    const int nb    = n0 + idx16;
    const bool nOk  = (nb < N_OUT);

    v8f acc = {};
    #pragma unroll
    for (int k0 = 0; k0 < HIDDEN; k0 += 4) {
        const int ka = k0 + half * 2;
        v2f a;
        a.x = As[idx16 * HIDDEN + ka];
        a.y = As[idx16 * HIDDEN + ka + 1];
        v2f b;
        b.x = nOk ? W2[(size_t)ka       * N_OUT + nb] : 0.f;
        b.y = nOk ? W2[(size_t)(ka + 1) * N_OUT + nb] : 0.f;
        acc = __builtin_amdgcn_wmma_f32_16x16x4_f32(false, a, false, b,
                                                    (short)0, acc, false, false);
    }
    if (nOk) {
        #pragma unroll
        for (int r = 0; r < 8; ++r) {
            const int m = m0 + r + half * 8;
            if (m < N) h2[(size_t)m * N_OUT + nb] = acc[r];
        }
    }
}

// ---------------- layer-2 edge aggregation into out ----------------
__global__ void agg2_kernel(const int* __restrict__ srcI, const int* __restrict__ dstI,
                            const float* __restrict__ dinv, const float* __restrict__ h2,
                            float* __restrict__ out, int E) {
    int lane = threadIdx.x & 31;
    int e = blockIdx.x * 8 + (threadIdx.x >> 5);
    if (e >= E || lane >= N_OUT) return;
    int s = srcI[e], d = dstI[e];
    float w = dinv[s] * dinv[d];
    atomicAdd(&out[(size_t)d * N_OUT + lane], h2[(size_t)s * N_OUT + lane] * w);
}

// ---------------- layer-2 self loop + bias ----------------
__global__ void final_kernel(const float* __restrict__ h2, const float* __restrict__ dinv,
                             const float* __restrict__ b2, float* __restrict__ out, int total) {
    int i = blockIdx.x * blockDim.x + threadIdx.x;
    if (i >= total) return;
    int n = i / N_OUT;
    int f = i - n * N_OUT;
    float dv = dinv[n];
    out[i] += h2[i] * dv * dv + b2[f];
}

extern "C" void kernel_launch(void* const* d_in, const int* in_sizes, int n_in,
                              void* d_out, int out_size, void* d_ws, size_t ws_size,
                              hipStream_t stream) {
    const float* x   = (const float*)d_in[0];
    const int*  eidx = (const int*)d_in[1];
    const float* W1  = (const float*)d_in[2];
    const float* b1  = (const float*)d_in[3];
    const float* W2  = (const float*)d_in[4];
    const float* b2  = (const float*)d_in[5];
    float* out = (float*)d_out;

    const int N = in_sizes[0] / N_FEAT;     // 100000
    const int E = in_sizes[1] / 2;          // 1600000
    const int* srcI = eidx;
    const int* dstI = eidx + E;

    float* ws   = (float*)d_ws;
    float* dinv = ws;                            // N
    float* h    = dinv + N;                      // N*64
    float* agg  = h + (size_t)N * HIDDEN;        // N*64 (reused as relu output)
    float* h2   = agg + (size_t)N * HIDDEN;      // N*21

    const int threads = 256;

    // init: deg=1 (self loop), agg=0, out=0
    fill_kernel<<<(N + threads - 1) / threads, threads, 0, stream>>>(dinv, 1.0f, N);
    {
        int tot = N * HIDDEN;
        fill_kernel<<<(tot + threads - 1) / threads, threads, 0, stream>>>(agg, 0.0f, tot);
    }
    fill_kernel<<<(out_size + threads - 1) / threads, threads, 0, stream>>>(out, 0.0f, out_size);

    // degrees and normalization
    deg_kernel<<<(E + threads - 1) / threads, threads, 0, stream>>>(dstI, dinv, E);
    rsqrt_kernel<<<(N + threads - 1) / threads, threads, 0, stream>>>(dinv, N);

    // layer 1
    gemm1_kernel<<<(N + 15) / 16, 128, 0, stream>>>(x, W1, h, N);
    {
        int tot = E * HIDDEN;  // 102,400,000 fits in int
        agg1_kernel<<<(tot + 255) / 256, 256, 0, stream>>>(srcI, dstI, dinv, h, agg, E);
    }
    {
        int tot = N * HIDDEN;
        sbr_kernel<<<(tot + threads - 1) / threads, threads, 0, stream>>>(h, dinv, b1, agg, tot);
    }

    // layer 2
    gemm2_kernel<<<(N + 15) / 16, 64, 0, stream>>>(agg, W2, h2, N);
    agg2_kernel<<<(E + 7) / 8, 256, 0, stream>>>(srcI, dstI, dinv, h2, out, E);
    final_kernel<<<(out_size + threads - 1) / threads, threads, 0, stream>>>(h2, dinv, b2, out, out_size);
}